// Net_18872086298709
// MI455X (gfx1250) — compile-verified
//
#include <hip/hip_runtime.h>

typedef float v2f __attribute__((ext_vector_type(2)));
typedef float v8f __attribute__((ext_vector_type(8)));

#define NSTEPS 10

__device__ __forceinline__ float fast_tanh(float x) {
#if __has_builtin(__builtin_amdgcn_tanhf)
  return __builtin_amdgcn_tanhf(x);
#else
  float r;
  asm volatile("v_tanh_f32 %0, %1" : "=v"(r) : "v"(x));
  return r;
#endif
}

// SWAPX16 across the two wave32 halves: xor=0x10, or=0x00, and=0x1f -> 0x401f
__device__ __forceinline__ float swap_half(float x) {
  return __int_as_float(__builtin_amdgcn_ds_swizzle(__float_as_int(x), 0x401f));
}

__global__ void write_tseq_kernel(float* __restrict__ out) {
  int i = threadIdx.x;
  if (i < 11) out[i] = (i < 10) ? (float)i * 0.01f : 0.1f;
}

__global__ __launch_bounds__(256) void node_rk4_wmma_kernel(
    const float* __restrict__ x, const float* __restrict__ W1,
    const float* __restrict__ b1, const float* __restrict__ W2,
    const float* __restrict__ b2, float* __restrict__ out, int nsamp) {
  const int lane = threadIdx.x & 31;
  const int wave = threadIdx.x >> 5;
  const int col  = lane & 15;            // sample column within the 16-wide tile
  const bool low = lane < 16;
  const int tile = blockIdx.x * (blockDim.x >> 5) + wave;
  if (tile * 16 >= nsamp) return;        // wave-uniform: EXEC stays all-ones
  const int sample = tile * 16 + col;
  const bool valid = sample < nsamp;
  const int li = valid ? sample : (nsamp - 1);

  // ---- A matrix: W1 (16x2) zero-padded to 16x4 in WMMA A-layout ----
  // lanes 0-15: VGPR0=K0, VGPR1=K1 for row M=lane; lanes 16-31: K2,K3 = 0
  v2f A;
  A.x = low ? W1[2 * col + 0] : 0.0f;
  A.y = low ? W1[2 * col + 1] : 0.0f;

  // ---- C matrix: b1 broadcast across columns, in C/D layout ----
  const int ro = low ? 0 : 8;
  v8f Cb;
  float w2a[8], w2b[8];
#pragma unroll
  for (int r = 0; r < 8; ++r) {
    Cb[r]  = b1[r + ro];
    w2a[r] = W2[r + ro];        // W2[0][r+ro]
    w2b[r] = W2[16 + r + ro];   // W2[1][r+ro]
  }
  const float b20 = b2[0], b21 = b2[1];

  float yx = x[2 * li + 0];
  float yy = x[2 * li + 1];

  float* traj = out + 11;  // [11][N][2] after t_seq
  if (low && valid) {
    size_t idx = (size_t)sample * 2;   // t = 0 plane
    traj[idx + 0] = yx;
    traj[idx + 1] = yy;
  }

  // f(u) = tanh(W1 u + b1) . W2 + b2, 16 samples per wave via one f32 WMMA
  auto F = [&](float ux, float uy, float& fx, float& fy) {
    v2f B;                 // B-layout 4x16: VGPR0 lanes0-15 = K0 row (= ux),
    B.x = ux;              // VGPR1 lanes0-15 = K1 row (= uy). High-lane K2/K3
    B.y = uy;              // contents multiply A's zero padding -> don't care.
    v8f h = __builtin_amdgcn_wmma_f32_16x16x4_f32(
        false, A, false, B, (short)0, Cb, false, false);
    float p0 = 0.0f, p1 = 0.0f;
#pragma unroll
    for (int r = 0; r < 8; ++r) {
      float t = fast_tanh(h[r]);
      p0 = fmaf(w2a[r], t, p0);
      p1 = fmaf(w2b[r], t, p1);
    }
    // rows 0-7 partial in low half, rows 8-15 partial in high half: combine
    p0 += swap_half(p0);
    p1 += swap_half(p1);
    fx = p0 + b20;
    fy = p1 + b21;
  };

  for (int s = 0; s < NSTEPS; ++s) {
    // replicate jnp.diff(concat(arange(0,0.1,0.01),[0.1])) exactly in f32
    float t0 = (float)s * 0.01f;
    float t1 = (s == NSTEPS - 1) ? 0.1f : (float)(s + 1) * 0.01f;
    float h = t1 - t0;

    float k1x, k1y, k2x, k2y, k3x, k3y, k4x, k4y;
    F(yx, yy, k1x, k1y);
    F(fmaf(0.5f * h, k1x, yx), fmaf(0.5f * h, k1y, yy), k2x, k2y);
    F(fmaf(0.5f * h, k2x, yx), fmaf(0.5f * h, k2y, yy), k3x, k3y);
    F(fmaf(h, k3x, yx), fmaf(h, k3y, yy), k4x, k4y);

    float c = h * (1.0f / 6.0f);
    yx += c * (k1x + 2.0f * (k2x + k3x) + k4x);
    yy += c * (k1y + 2.0f * (k2y + k3y) + k4y);

    if (low && valid) {
      size_t idx = ((size_t)(s + 1) * (size_t)nsamp + (size_t)sample) * 2;
      traj[idx + 0] = yx;
      traj[idx + 1] = yy;
    }
  }
}

extern "C" void kernel_launch(void* const* d_in, const int* in_sizes, int n_in,
                              void* d_out, int out_size, void* d_ws, size_t ws_size,
                              hipStream_t stream) {
  const float* x  = (const float*)d_in[0];
  const float* W1 = (const float*)d_in[1];
  const float* b1 = (const float*)d_in[2];
  const float* W2 = (const float*)d_in[3];
  const float* b2 = (const float*)d_in[4];
  float* out = (float*)d_out;
  const int nsamp = in_sizes[0] / 2;

  write_tseq_kernel<<<1, 32, 0, stream>>>(out);

  // 256 threads = 8 waves = 128 samples per block
  const int samples_per_block = 128;
  const int blocks = (nsamp + samples_per_block - 1) / samples_per_block;
  node_rk4_wmma_kernel<<<blocks, 256, 0, stream>>>(x, W1, b1, W2, b2, out, nsamp);
}